// SAGEBlock_35115652612241
// MI455X (gfx1250) — compile-verified
//
#include <hip/hip_runtime.h>
#include <stdint.h>

#define IN_DIM   128
#define OUT_DIM  128
#define N_NODES  50000
#define N_EDGES  800000

typedef __attribute__((ext_vector_type(2))) float v2f;
typedef __attribute__((ext_vector_type(8))) float v8f;

// ---------------------------------------------------------------------------
// Kernel 1: zero the workspace (agg accumulator + degree counts).
// ---------------------------------------------------------------------------
__global__ void sage_zero_ws(float* __restrict__ agg, float* __restrict__ cnt) {
  const long long stride = (long long)gridDim.x * blockDim.x;
  const long long tid    = (long long)blockIdx.x * blockDim.x + threadIdx.x;
  const long long tot    = (long long)N_NODES * IN_DIM;
  for (long long i = tid; i < tot; i += stride) agg[i] = 0.0f;
  for (long long i = tid; i < N_NODES; i += stride) cnt[i] = 0.0f;
}

// ---------------------------------------------------------------------------
// Kernel 2: edge scatter. One wave32 per edge (grid-stride over edges).
// Each lane gathers 4 contiguous features of x[src] (float4 -> b128 load)
// and issues 4 non-returning f32 atomics into agg[dst]. Lane 0 counts degree.
// ---------------------------------------------------------------------------
__global__ __launch_bounds__(256)
void sage_scatter(const float* __restrict__ x,
                  const long long* __restrict__ ei,   // [2, N_EDGES] int64
                  float* __restrict__ agg,
                  float* __restrict__ cnt) {
  const int lane   = threadIdx.x & 31;
  const int wid    = (int)((blockIdx.x * blockDim.x + threadIdx.x) >> 5);
  const int nwaves = (int)((gridDim.x * blockDim.x) >> 5);
  for (int e = wid; e < N_EDGES; e += nwaves) {
    // prefetch the edge list ahead of us (global_prefetch_b8)
    if (e + nwaves < N_EDGES) {
      __builtin_prefetch(ei + e + nwaves, 0, 0);
      __builtin_prefetch(ei + N_EDGES + e + nwaves, 0, 0);
    }
    const int src = (int)ei[e];
    const int dst = (int)ei[N_EDGES + e];
    const float4 v = ((const float4*)(x + (long long)src * IN_DIM))[lane];
    float* p = agg + (long long)dst * IN_DIM + lane * 4;
    atomicAdd(p + 0, v.x);
    atomicAdd(p + 1, v.y);
    atomicAdd(p + 2, v.z);
    atomicAdd(p + 3, v.w);
    if (lane == 0) atomicAdd(cnt + dst, 1.0f);
  }
}

// ---------------------------------------------------------------------------
// Kernel 3: agg /= max(cnt, 1)  (mean aggregation), float4-vectorized.
// ---------------------------------------------------------------------------
__global__ void sage_finalize_mean(float* __restrict__ agg,
                                   const float* __restrict__ cnt) {
  const int i = blockIdx.x * blockDim.x + threadIdx.x;      // one float4 each
  const int tot4 = N_NODES * (IN_DIM / 4);
  if (i >= tot4) return;
  const int node = i >> 5;                                  // 32 float4 / row
  const float inv = 1.0f / fmaxf(cnt[node], 1.0f);
  float4* p = ((float4*)agg) + i;
  float4 v = *p;
  v.x *= inv; v.y *= inv; v.z *= inv; v.w *= inv;
  *p = v;
}

// ---------------------------------------------------------------------------
// CDNA5 async copy: global -> LDS, 16 bytes per lane, tracked by ASYNCcnt.
// lds_off: LDS byte address (low 32 bits of flat shared pointer, ISA 10.2),
// byte_off: per-lane unsigned byte offset added to the SGPR base (GVS mode).
// ---------------------------------------------------------------------------
__device__ __forceinline__ void async_copy_b128(uint32_t lds_off,
                                                const float* gbase,
                                                uint32_t byte_off) {
  asm volatile("global_load_async_to_lds_b128 %0, %1, %2"
               :: "v"(lds_off), "v"(byte_off), "s"(gbase)
               : "memory");
}

__device__ __forceinline__ void wait_asynccnt0() {
  asm volatile("s_wait_asynccnt 0x0" ::: "memory");
}

// ---------------------------------------------------------------------------
// Kernel 4: fused  out = LN(agg@W_l + b_l + x@W_r) -> ReLU  via WMMA f32.
//
// Block = 16 output rows x 128 cols. 8 waves, wave w owns cols [16w,16w+16).
// W_l / W_r (64 KB each) are staged into LDS once per block with
// global_load_async_to_lds_b128; B fragments then come from LDS
// (ds_load_2addr, 512 B apart). K=128 -> 32 steps of V_WMMA_F32_16X16X4_F32,
// two per step (W_l and W_r paths accumulate into the same v8f C).
//
// 16x4 f32 A layout (ISA 7.12.2): lane m = lane&15, K-half = lane>>4,
//   VGPR0 = K(2*kh), VGPR1 = K(2*kh+1)  -> contiguous float2 per lane.
// 4x16 B layout mirrors it with n = lane&15 along N.
// 16x16 f32 C layout: c[v] = row (v + 8*(lane>>4)), col (lane&15).
// ---------------------------------------------------------------------------
__global__ __launch_bounds__(256)
void sage_wmma_ln_relu(const float* __restrict__ x,
                       const float* __restrict__ agg,
                       const float* __restrict__ Wl,
                       const float* __restrict__ bl,
                       const float* __restrict__ Wr,
                       const float* __restrict__ gamma,
                       const float* __restrict__ beta,
                       float* __restrict__ out) {
  __shared__ float wl_lds[IN_DIM * OUT_DIM];   // 64 KB
  __shared__ float wr_lds[IN_DIM * OUT_DIM];   // 64 KB
  __shared__ float tile[16 * 128];             // 8 KB
  __shared__ float mu_s[16];
  __shared__ float rs_s[16];

  const int tid  = threadIdx.x;
  const int lane = tid & 31;
  const int wave = tid >> 5;
  const int n    = lane & 15;     // A: row-in-tile (M); B/C: col-in-tile (N)
  const int kh   = lane >> 4;     // K half-select (0/1)
  const int c0   = wave * 16;     // this wave's output column base

  // ---- Stage W_l | W_r into LDS (async, 16B per lane per op) ----
  for (int i = tid; i < (IN_DIM * OUT_DIM) / 4; i += 256) {
    const uint32_t boff = (uint32_t)i * 16u;
    async_copy_b128((uint32_t)(uintptr_t)&wl_lds[i * 4], Wl, boff);
    async_copy_b128((uint32_t)(uintptr_t)&wr_lds[i * 4], Wr, boff);
  }
  wait_asynccnt0();
  __syncthreads();

  const long long row0 = (long long)blockIdx.x * 16;
  const float* aggRow = agg + (row0 + n) * IN_DIM;
  const float* xRow   = x   + (row0 + n) * IN_DIM;

  v8f c = {};
  #pragma unroll 4
  for (int k0 = 0; k0 < IN_DIM; k0 += 4) {
    const int ka = k0 + 2 * kh;
    v2f aL = { aggRow[ka], aggRow[ka + 1] };
    v2f aX = { xRow[ka],   xRow[ka + 1]   };
    v2f bL = { wl_lds[ka * OUT_DIM + c0 + n], wl_lds[(ka + 1) * OUT_DIM + c0 + n] };
    v2f bR = { wr_lds[ka * OUT_DIM + c0 + n], wr_lds[(ka + 1) * OUT_DIM + c0 + n] };
    c = __builtin_amdgcn_wmma_f32_16x16x4_f32(false, aL, false, bL,
                                              (short)0, c, false, false);
    c = __builtin_amdgcn_wmma_f32_16x16x4_f32(false, aX, false, bR,
                                              (short)0, c, false, false);
  }

  // Epilogue part 1: bias + spill C tile to LDS for the row-wise LayerNorm.
  const float bias = bl[c0 + n];
  #pragma unroll
  for (int v = 0; v < 8; ++v) {
    const int r = v + 8 * kh;
    tile[r * 128 + c0 + n] = c[v] + bias;
  }
  __syncthreads();

  // Epilogue part 2: per-row mean / variance (16 rows, one thread each).
  if (tid < 16) {
    float s = 0.0f, s2 = 0.0f;
    #pragma unroll 8
    for (int j = 0; j < 128; ++j) {
      const float t = tile[tid * 128 + j];
      s  += t;
      s2 += t * t;
    }
    const float mu  = s * (1.0f / 128.0f);
    const float var = s2 * (1.0f / 128.0f) - mu * mu;
    mu_s[tid] = mu;
    rs_s[tid] = rsqrtf(var + 1e-5f);
  }
  __syncthreads();

  // Epilogue part 3: normalize, affine, ReLU; 8 contiguous floats / thread.
  {
    const int r   = tid >> 4;
    const int col = (tid & 15) * 8;
    const float mu = mu_s[r];
    const float rs = rs_s[r];
    float* op = out + (row0 + r) * OUT_DIM + col;
    #pragma unroll
    for (int j = 0; j < 8; ++j) {
      const float v = (tile[r * 128 + col + j] - mu) * rs * gamma[col + j]
                      + beta[col + j];
      op[j] = fmaxf(v, 0.0f);
    }
  }
}

// ---------------------------------------------------------------------------
// Host-side launcher.
// Inputs (setup_inputs order): x, edge_index(int64), W_l, b_l, W_r, gamma, beta
// Workspace layout: agg [N_NODES*IN_DIM f32] | cnt [N_NODES f32]  (~25.8 MB)
// ---------------------------------------------------------------------------
extern "C" void kernel_launch(void* const* d_in, const int* in_sizes, int n_in,
                              void* d_out, int out_size, void* d_ws, size_t ws_size,
                              hipStream_t stream) {
  (void)in_sizes; (void)n_in; (void)out_size; (void)ws_size;

  const float*     x     = (const float*)d_in[0];
  const long long* ei    = (const long long*)d_in[1];
  const float*     Wl    = (const float*)d_in[2];
  const float*     bl    = (const float*)d_in[3];
  const float*     Wr    = (const float*)d_in[4];
  const float*     gamma = (const float*)d_in[5];
  const float*     beta  = (const float*)d_in[6];
  float*           out   = (float*)d_out;

  float* agg = (float*)d_ws;
  float* cnt = agg + (size_t)N_NODES * IN_DIM;

  sage_zero_ws<<<1024, 256, 0, stream>>>(agg, cnt);

  // 4096 blocks * 8 waves = 32768 waves grid-striding 800K edges.
  sage_scatter<<<4096, 256, 0, stream>>>(x, ei, agg, cnt);

  const int tot4 = N_NODES * (IN_DIM / 4);
  sage_finalize_mean<<<(tot4 + 255) / 256, 256, 0, stream>>>(agg, cnt);

  // 50000 rows / 16 rows-per-block = 3125 blocks, 8 waves each.
  sage_wmma_ln_relu<<<N_NODES / 16, 256, 0, stream>>>(
      x, agg, Wl, bl, Wr, gamma, beta, out);
}